// MLPPredictor_65103114273054
// MI455X (gfx1250) — compile-verified
//
#include <hip/hip_runtime.h>

typedef __attribute__((ext_vector_type(2))) float v2f;
typedef __attribute__((ext_vector_type(8))) float v8f;

#define IN_FEAT     64
#define EDGE_FEAT   32
#define OUT_CLASSES 8
#define FAN_IN      (2 * IN_FEAT + EDGE_FEAT)   // 160
#define WAVES_PER_BLOCK 8                        // 256 threads = 8 wave32

// ---------------------------------------------------------------------------
// Phase 1: per-node projections with V_WMMA_F32_16X16X4_F32.
//   PQ[n][0:8]  = h[n,:] @ W[:, 0:64]^T   (P)
//   PQ[n][8:16] = h[n,:] @ W[:,64:128]^T  (Q)
// One wave handles a 16-node tile; B packs W1 in cols 0-7 and W2 in cols 8-15
// so a single K=64 chain (16 WMMAs) yields both P and Q.
// ---------------------------------------------------------------------------
__global__ __launch_bounds__(256) void node_proj_kernel(
    const float* __restrict__ h, const float* __restrict__ W,
    float* __restrict__ PQ, int nTiles, int nNodes)
{
    const int lane = threadIdx.x & 31;
    const int tile = blockIdx.x * WAVES_PER_BLOCK + (threadIdx.x >> 5);
    if (tile >= nTiles) return;                 // wave-uniform guard

    const int m  = lane & 15;                   // A row within tile / B column n
    const int kg = lane >> 4;                   // K-group (0 or 1)
    const int n  = m;                           // D column this lane owns
    const int cls = n & 7;
    const int woff = (n < 8) ? 0 : IN_FEAT;     // W1 vs W2 slice

    const int nodeA = min(tile * 16 + m, nNodes - 1);   // clamped A row
    const float* hrow = h + (size_t)nodeA * IN_FEAT;
    const float* wrow = W + (size_t)cls * FAN_IN + woff;

    v8f c = {};
    #pragma unroll
    for (int s = 0; s < 16; ++s) {
        const int kk = s * 4 + 2 * kg;
        // A 16x4: lane (m,kg) holds A[m][kk], A[m][kk+1]  -> contiguous b64 load
        v2f a = *(const v2f*)(hrow + kk);
        // B 4x16: lane (n,kg) holds B[kk][n], B[kk+1][n] = W[cls][woff+kk..]
        v2f b = *(const v2f*)(wrow + kk);
        c = __builtin_amdgcn_wmma_f32_16x16x4_f32(
                false, a, false, b, (short)0, c, false, false);
    }

    // D layout: VGPR v, lane l -> element (M = v + 8*kg, N = n)
    #pragma unroll
    for (int v = 0; v < 8; ++v) {
        const int row = tile * 16 + v + 8 * kg;     // PQ padded to nTiles*16 rows
        PQ[(size_t)row * 16 + n] = c[v];
    }
}

// ---------------------------------------------------------------------------
// Phase 2: per-edge scores. C seeded with b[n] + P[src][n] + Q[dst][n]
// (8-float L2-resident gathers), then 8 WMMAs accumulate edge_attr @ W3^T.
// ---------------------------------------------------------------------------
__global__ __launch_bounds__(256) void edge_score_kernel(
    const float* __restrict__ edge_attr,
    const int*   __restrict__ src, const int* __restrict__ dst,
    const float* __restrict__ W, const float* __restrict__ bias,
    const float* __restrict__ PQ,
    float* __restrict__ out, int nTiles, int nEdges)
{
    const int lane = threadIdx.x & 31;
    const int tile = blockIdx.x * WAVES_PER_BLOCK + (threadIdx.x >> 5);
    if (tile >= nTiles) return;                 // wave-uniform guard

    const int m  = lane & 15;
    const int kg = lane >> 4;
    const int n  = m;
    const int cls = n & 7;

    // Seed C with bias + gathered node projections (divergent region re-converges
    // before any WMMA; EXEC is all-ones at every WMMA issue).
    v8f c;
    #pragma unroll
    for (int v = 0; v < 8; ++v) {
        float acc = 0.0f;
        if (n < 8) {
            const int e = min(tile * 16 + v + 8 * kg, nEdges - 1);
            const int s = src[e];
            const int d = dst[e];
            acc = bias[n] + PQ[(size_t)s * 16 + n] + PQ[(size_t)d * 16 + 8 + n];
        }
        c[v] = acc;
    }

    const int eA = min(tile * 16 + m, nEdges - 1);
    const float* erow = edge_attr + (size_t)eA * EDGE_FEAT;
    const float* wrow = W + (size_t)cls * FAN_IN + 2 * IN_FEAT;   // W3 slice

    #pragma unroll
    for (int s = 0; s < 8; ++s) {
        const int kk = s * 4 + 2 * kg;
        v2f a = *(const v2f*)(erow + kk);
        v2f b;
        if (n < 8) b = *(const v2f*)(wrow + kk);
        else       { b.x = 0.0f; b.y = 0.0f; }      // pad columns contribute 0
        c = __builtin_amdgcn_wmma_f32_16x16x4_f32(
                false, a, false, b, (short)0, c, false, false);
    }

    #pragma unroll
    for (int v = 0; v < 8; ++v) {
        const int e = tile * 16 + v + 8 * kg;
        if (n < 8 && e < nEdges)
            out[(size_t)e * OUT_CLASSES + n] = c[v];
    }
}

// ---------------------------------------------------------------------------
// Fallback: fully fused single pass (used only if workspace is too small).
// K = 160 = 40 WMMA steps; feature segment per step is a compile-time choice.
// ---------------------------------------------------------------------------
__global__ __launch_bounds__(256) void fused_edge_kernel(
    const float* __restrict__ h, const float* __restrict__ edge_attr,
    const int*   __restrict__ src, const int* __restrict__ dst,
    const float* __restrict__ W, const float* __restrict__ bias,
    float* __restrict__ out, int nTiles, int nEdges)
{
    const int lane = threadIdx.x & 31;
    const int tile = blockIdx.x * WAVES_PER_BLOCK + (threadIdx.x >> 5);
    if (tile >= nTiles) return;

    const int m  = lane & 15;
    const int kg = lane >> 4;
    const int n  = m;
    const int cls = n & 7;

    const int eA = min(tile * 16 + m, nEdges - 1);
    const int se = src[eA];
    const int de = dst[eA];
    const float* hu = h + (size_t)se * IN_FEAT;
    const float* hv = h + (size_t)de * IN_FEAT;
    const float* er = edge_attr + (size_t)eA * EDGE_FEAT;
    const float* wrow = W + (size_t)cls * FAN_IN;

    v8f c;
    #pragma unroll
    for (int v = 0; v < 8; ++v)
        c[v] = (n < 8) ? bias[n] : 0.0f;

    #pragma unroll
    for (int s = 0; s < 40; ++s) {
        const int k0 = s * 4;
        const int kk = k0 + 2 * kg;
        v2f a;
        if (k0 < IN_FEAT)            a = *(const v2f*)(hu + kk);
        else if (k0 < 2 * IN_FEAT)   a = *(const v2f*)(hv + (kk - IN_FEAT));
        else                         a = *(const v2f*)(er + (kk - 2 * IN_FEAT));
        v2f b;
        if (n < 8) b = *(const v2f*)(wrow + kk);
        else       { b.x = 0.0f; b.y = 0.0f; }
        c = __builtin_amdgcn_wmma_f32_16x16x4_f32(
                false, a, false, b, (short)0, c, false, false);
    }

    #pragma unroll
    for (int v = 0; v < 8; ++v) {
        const int e = tile * 16 + v + 8 * kg;
        if (n < 8 && e < nEdges)
            out[(size_t)e * OUT_CLASSES + n] = c[v];
    }
}

extern "C" void kernel_launch(void* const* d_in, const int* in_sizes, int n_in,
                              void* d_out, int out_size, void* d_ws, size_t ws_size,
                              hipStream_t stream)
{
    const float* h    = (const float*)d_in[0];
    const float* ea   = (const float*)d_in[1];
    const int*   src  = (const int*)d_in[2];
    const int*   dst  = (const int*)d_in[3];
    const float* W    = (const float*)d_in[4];
    const float* bias = (const float*)d_in[5];
    float* out = (float*)d_out;

    const int nNodes = in_sizes[0] / IN_FEAT;
    const int nEdges = in_sizes[2];
    const int nodeTiles = (nNodes + 15) / 16;
    const int edgeTiles = (nEdges + 15) / 16;

    const size_t pqBytes = (size_t)nodeTiles * 16 * 16 * sizeof(float);

    if (ws_size >= pqBytes) {
        float* PQ = (float*)d_ws;
        node_proj_kernel<<<(nodeTiles + WAVES_PER_BLOCK - 1) / WAVES_PER_BLOCK,
                           256, 0, stream>>>(h, W, PQ, nodeTiles, nNodes);
        edge_score_kernel<<<(edgeTiles + WAVES_PER_BLOCK - 1) / WAVES_PER_BLOCK,
                            256, 0, stream>>>(ea, src, dst, W, bias, PQ, out,
                                              edgeTiles, nEdges);
    } else {
        fused_edge_kernel<<<(edgeTiles + WAVES_PER_BLOCK - 1) / WAVES_PER_BLOCK,
                            256, 0, stream>>>(h, ea, src, dst, W, bias, out,
                                              edgeTiles, nEdges);
    }
}